// WSGATLayer_3186865734208
// MI455X (gfx1250) — compile-verified
//
#include <hip/hip_runtime.h>
#include <hip/hip_bf16.h>

typedef __attribute__((ext_vector_type(2))) float v2f;
typedef __attribute__((ext_vector_type(8))) float v8f;

#define NEG_SLOPE 0.01f
#define IN_DIM 128
#define OUT_DIM 64
#define FEAT 50

// ---- helpers -------------------------------------------------------------

__device__ __forceinline__ float wave_sum(float v) {
  #pragma unroll
  for (int off = 16; off > 0; off >>= 1) v += __shfl_xor(v, off, 32);
  return v;
}

// monotonic float<->uint encoding so atomicMax(uint) == max(float)
__device__ __forceinline__ unsigned enc_f32(float f) {
  unsigned u = __float_as_uint(f);
  return (u & 0x80000000u) ? ~u : (u | 0x80000000u);
}
__device__ __forceinline__ float dec_f32(unsigned k) {
  unsigned u = (k & 0x80000000u) ? (k ^ 0x80000000u) : ~k;
  return __uint_as_float(u);
}

// ---- tiny: wf[k] = sum_d W_feat[d][k] * wa[2D+d] -------------------------

__global__ void k_wf(const float* __restrict__ Wfeat,
                     const float* __restrict__ Wattn,
                     float* __restrict__ wf) {
  int k = threadIdx.x;
  if (k >= FEAT) return;
  float s = 0.f;
  #pragma unroll
  for (int d = 0; d < OUT_DIM; ++d) s += Wfeat[d * FEAT + k] * Wattn[2 * OUT_DIM + d];
  wf[k] = s;
}

// ---- z = h @ W_fc.T via V_WMMA_F32_16X16X4_F32 ---------------------------
// one wave per 16x16 output tile; K=128 -> 32 wmma steps

__global__ void k_gemm_z(const float* __restrict__ h,
                         const float* __restrict__ Wfc,
                         float* __restrict__ z, int n_total) {
  int wave = blockIdx.x * (blockDim.x >> 5) + (threadIdx.x >> 5);
  int lane = threadIdx.x & 31;
  int mt = wave >> 2;            // OUT_DIM/16 == 4 tiles in N
  int nt = wave & 3;
  int m0 = mt * 16;
  if (m0 >= n_total) return;
  int n0 = nt * 16;
  int half = lane >> 4;
  int lanelo = lane & 15;
  // A: lane holds row m0+lanelo, K = 2*half + {0,1}
  const float* arow = h   + (size_t)(m0 + lanelo) * IN_DIM + 2 * half;
  // B[k][n] = W_fc[n][k]; lane holds col n0+lanelo, K = 2*half + {0,1}
  const float* brow = Wfc + (size_t)(n0 + lanelo) * IN_DIM + 2 * half;
  v8f c = {};
  #pragma unroll 8
  for (int k = 0; k < IN_DIM; k += 4) {
    v2f A, B;
    A.x = arow[k]; A.y = arow[k + 1];
    B.x = brow[k]; B.y = brow[k + 1];
    c = __builtin_amdgcn_wmma_f32_16x16x4_f32(false, A, false, B,
                                              (short)0, c, false, false);
  }
  // D layout: VGPR r, lane-half -> M = r + 8*half, N = lanelo
  #pragma unroll
  for (int r = 0; r < 8; ++r) {
    int row = m0 + r + 8 * half;
    z[(size_t)row * OUT_DIM + n0 + lanelo] = c[r];
  }
}

// ---- per-node attention scalar: a[i] = z[i] . wa(role) -------------------

__global__ void k_attn_node(const float* __restrict__ z,
                            const float* __restrict__ Wattn,
                            float* __restrict__ a, int n_total, int n_words) {
  int node = blockIdx.x * (blockDim.x >> 5) + (threadIdx.x >> 5);
  int lane = threadIdx.x & 31;
  if (node >= n_total) return;
  const float* wa = Wattn + ((node < n_words) ? 0 : OUT_DIM);
  const float* zr = z + (size_t)node * OUT_DIM;
  float s = zr[lane] * wa[lane] + zr[lane + 32] * wa[lane + 32];
  s = wave_sum(s);
  if (lane == 0) a[node] = s;
}

// ---- zero the accumulators (ws is poisoned 0xAA by harness) --------------

__global__ void k_init(unsigned* __restrict__ mkeys, float* __restrict__ denom,
                       float* __restrict__ shacc, int n_sents) {
  int tid = blockIdx.x * blockDim.x + threadIdx.x;
  if (tid < n_sents * OUT_DIM) shacc[tid] = 0.f;
  if (tid < n_sents) { mkeys[tid] = 0u; denom[tid] = 0.f; }
}

// ---- pass A: e = leaky(a[src] + a[nw+dst] + tfidf.wf); segment max -------
// one wave per edge: coalesced 200B tfidf stream per wave

__global__ void k_edge_a(const float* __restrict__ tfidf,
                         const float* __restrict__ wf,
                         const float* __restrict__ a,
                         const int* __restrict__ src,
                         const int* __restrict__ dst,
                         float* __restrict__ e_arr,
                         unsigned* __restrict__ mkeys,
                         int E, int n_words) {
  int edge = blockIdx.x * (blockDim.x >> 5) + (threadIdx.x >> 5);
  int lane = threadIdx.x & 31;
  if (edge >= E) return;
  const float* tr = tfidf + (size_t)edge * FEAT;
  // keep the dominant 400MB stream ahead of the wave
  __builtin_prefetch(tr + 8 * FEAT, 0, 0);
  float s = 0.f;
  s += tr[lane] * wf[lane];                       // k = 0..31
  if (lane < FEAT - 32) s += tr[lane + 32] * wf[lane + 32];  // k = 32..49
  s = wave_sum(s);
  if (lane == 0) {
    float ev = a[src[edge]] + a[n_words + dst[edge]] + s;
    ev = (ev >= 0.f) ? ev : NEG_SLOPE * ev;
    e_arr[edge] = ev;
    atomicMax(&mkeys[dst[edge]], enc_f32(ev));
  }
}

// ---- pass B: ex = exp(e - m[dst]); denom += ex; shacc += ex * z[src] -----

__global__ void k_edge_b(const float* __restrict__ z,
                         const float* __restrict__ e_arr,
                         const unsigned* __restrict__ mkeys,
                         const int* __restrict__ src,
                         const int* __restrict__ dst,
                         float* __restrict__ denom,
                         float* __restrict__ shacc, int E) {
  int edge = blockIdx.x * (blockDim.x >> 5) + (threadIdx.x >> 5);
  int lane = threadIdx.x & 31;
  if (edge >= E) return;
  int si = src[edge], di = dst[edge];
  float m = dec_f32(mkeys[di]);
  float ex = __expf(e_arr[edge] - m);
  if (lane == 0) atomicAdd(&denom[di], ex);
  const float* zr = z + (size_t)si * OUT_DIM;
  float* sa = shacc + (size_t)di * OUT_DIM;
  atomicAdd(&sa[lane],      ex * zr[lane]);
  atomicAdd(&sa[lane + 32], ex * zr[lane + 32]);
}

// ---- pass C: sh = shacc / denom (0 for empty segments) -------------------

__global__ void k_final(const float* __restrict__ shacc,
                        const float* __restrict__ denom,
                        float* __restrict__ out, int n_sents) {
  int tid = blockIdx.x * blockDim.x + threadIdx.x;
  if (tid >= n_sents * OUT_DIM) return;
  float dn = denom[tid >> 6];
  out[tid] = (dn > 0.f) ? shacc[tid] / dn : 0.f;
}

// ---- launch --------------------------------------------------------------

extern "C" void kernel_launch(void* const* d_in, const int* in_sizes, int n_in,
                              void* d_out, int out_size, void* d_ws, size_t ws_size,
                              hipStream_t stream) {
  const float* h     = (const float*)d_in[0];
  const float* tfidf = (const float*)d_in[1];
  const float* Wfc   = (const float*)d_in[2];
  const float* Wfeat = (const float*)d_in[3];
  const float* Wattn = (const float*)d_in[4];
  const int*   src   = (const int*)d_in[5];
  const int*   dst   = (const int*)d_in[6];
  float* out = (float*)d_out;

  const int n_total = in_sizes[0] / IN_DIM;       // 220000
  const int E       = in_sizes[5];                // 2,000,000
  const int n_sents = out_size / OUT_DIM;         // 20,000
  const int n_words = n_total - n_sents;          // 200,000

  // workspace layout (floats)
  float* ws = (float*)d_ws;
  float*    z     = ws;                                   // n_total*64
  float*    a     = z + (size_t)n_total * OUT_DIM;        // n_total
  float*    wf    = a + n_total;                          // 64 (50 used)
  float*    e_arr = wf + 64;                              // E
  unsigned* mkeys = (unsigned*)(e_arr + E);               // n_sents
  float*    denom = (float*)(mkeys + n_sents);            // n_sents
  float*    shacc = denom + n_sents;                      // n_sents*64

  const int THREADS = 256;
  const int WPB = THREADS / 32;  // 8 waves per block (wave32)

  // 1) fold W_feat with wa3
  k_wf<<<1, 64, 0, stream>>>(Wfeat, Wattn, wf);

  // 2) z = h @ W_fc.T  (WMMA)
  {
    int tiles = (n_total / 16) * (OUT_DIM / 16);
    k_gemm_z<<<(tiles + WPB - 1) / WPB, THREADS, 0, stream>>>(h, Wfc, z, n_total);
  }

  // 3) per-node attention scalars
  k_attn_node<<<(n_total + WPB - 1) / WPB, THREADS, 0, stream>>>(z, Wattn, a,
                                                                 n_total, n_words);
  // 4) zero accumulators
  {
    int n = n_sents * OUT_DIM;
    k_init<<<(n + THREADS - 1) / THREADS, THREADS, 0, stream>>>(mkeys, denom,
                                                                shacc, n_sents);
  }
  // 5) edge pass A: scores + segment max
  k_edge_a<<<(E + WPB - 1) / WPB, THREADS, 0, stream>>>(tfidf, wf, a, src, dst,
                                                        e_arr, mkeys, E, n_words);
  // 6) edge pass B: exp, denom, weighted scatter
  k_edge_b<<<(E + WPB - 1) / WPB, THREADS, 0, stream>>>(z, e_arr, mkeys, src, dst,
                                                        denom, shacc, E);
  // 7) normalize
  {
    int n = n_sents * OUT_DIM;
    k_final<<<(n + THREADS - 1) / THREADS, THREADS, 0, stream>>>(shacc, denom,
                                                                 out, n_sents);
  }
}